// MSDeformAttn_19473381720293
// MI455X (gfx1250) — compile-verified
//
#include <hip/hip_runtime.h>
#include <hip/hip_bf16.h>

#define D_MODEL   256
#define N_HEADS   8
#define N_LEVELS  4
#define N_POINTS  4
#define D_HEAD    32
#define NBATCH    4
#define HLP       128   /* N_HEADS * N_LEVELS * N_POINTS */

typedef __attribute__((ext_vector_type(2))) float v2f;
typedef __attribute__((ext_vector_type(8))) float v8f;

// ---------------------------------------------------------------------------
// Weight transpose: W (K x N, row-major) -> Wt (N x K, row-major).
// One element per thread; reads coalesced. Tiny (<=64K elements), runs once.
// ---------------------------------------------------------------------------
template<int K, int N>
__global__ void transpose_kn(const float* __restrict__ W, float* __restrict__ Wt) {
    const int idx = blockIdx.x * 256 + threadIdx.x;   // over K*N
    const int k = idx / N;
    const int n = idx - k * N;
    Wt[(size_t)n * K + k] = W[idx];
}

// ---------------------------------------------------------------------------
// C[M,N] = A[M,K] @ W[K,N] + bias[N], with W pre-transposed (Wt is N x K).
//
// Per-wave register tile: 32 (M) x 64 (N) = 2 x 4 subtiles of 16x16 via
// V_WMMA_F32_16X16X4_F32.  Each K-step (K += 4):
//   - 2 A fragments : aligned v2f (global_load_b64)
//   - 4 B fragments : aligned v2f (global_load_b64)  [contiguous thanks to Wt]
//   - 8 WMMAs
// Block = 128 threads = 4 waves stacked in M (128 rows per block).
//
// fp32 WMMA operand layouts (CDNA5 ISA 7.12.2):
//  A (16x4): lane holds row M=lane%16; half=lane/16 selects K pair {0,1}|{2,3}
//  B (4x16): lane holds col N=lane%16; same K-pair selection by half
//  C/D     : lane holds col N=lane%16; VGPR v holds row M = v + 8*half
// ---------------------------------------------------------------------------
template<int KDIM, int NCOLS>
__global__ void wmma_gemm_bias(const float* __restrict__ A,
                               const float* __restrict__ Wt,   // NCOLS x KDIM
                               const float* __restrict__ bias,
                               float* __restrict__ C) {
    const int lane = threadIdx.x & 31;
    const int wid  = threadIdx.x >> 5;        // 0..3 : M slab within block
    const int half = lane >> 4;               // K-pair select
    const int lm   = lane & 15;

    const int m0 = (blockIdx.x * 4 + wid) * 32;   // 32 rows per wave
    const int n0 = blockIdx.y * 64;               // 64 cols per wave
    const int colB = n0 + lm;

    v8f acc[2][4];
#pragma unroll
    for (int t = 0; t < 4; ++t) {
        const float bv = bias[colB + 16 * t];
#pragma unroll
        for (int i = 0; i < 8; ++i) { acc[0][t][i] = bv; acc[1][t][i] = bv; }
    }

    const float* __restrict__ Arow0 = A  + (size_t)(m0      + lm) * KDIM + 2 * half;
    const float* __restrict__ Arow1 = A  + (size_t)(m0 + 16 + lm) * KDIM + 2 * half;
    const float* __restrict__ Wrow  = Wt + (size_t)colB * KDIM + 2 * half;

#pragma unroll 4
    for (int k = 0; k < KDIM; k += 4) {
        const v2f a0 = *(const v2f*)(Arow0 + k);          // 8B aligned
        const v2f a1 = *(const v2f*)(Arow1 + k);
#pragma unroll
        for (int t = 0; t < 4; ++t) {
            // b = (W[kk][col+16t], W[kk+1][col+16t]) : contiguous in Wt
            const v2f b = *(const v2f*)(Wrow + (size_t)(16 * t) * KDIM + k);
            acc[0][t] = __builtin_amdgcn_wmma_f32_16x16x4_f32(
                false, a0, false, b, (short)0, acc[0][t], false, false);
            acc[1][t] = __builtin_amdgcn_wmma_f32_16x16x4_f32(
                false, a1, false, b, (short)0, acc[1][t], false, false);
        }
    }

#pragma unroll
    for (int s = 0; s < 2; ++s) {
        float* __restrict__ Crow = C + (size_t)(m0 + 16 * s + 8 * half) * NCOLS + colB;
#pragma unroll
        for (int t = 0; t < 4; ++t)
#pragma unroll
            for (int v = 0; v < 8; ++v)
                Crow[(size_t)v * NCOLS + 16 * t] = acc[s][t][v];
    }
}

// ---------------------------------------------------------------------------
// Deformable sampling: one wave per (n, q, m); lane = channel d (Dh = 32).
//   value: (N, S, 256)  fp32   (head-major inner: m*32 + d)
//   refp : (N, Lq, L, 1)
//   off  : (N, Lq, M, L, P) = (N, Lq, 128)
//   awlog: (N, Lq, M, L*P)  — softmax over the 16 (l,p) logits per head
//   attn : (N, Lq, 256) output accumulator
// Each (l,p) gather touches one 128-byte line of `value` (fully coalesced);
// the 31 MB value tensor is L2-resident (192 MB L2).
// ---------------------------------------------------------------------------
__global__ void msdeform_sample(const float* __restrict__ value,
                                const float* __restrict__ refp,
                                const float* __restrict__ off,
                                const float* __restrict__ awlog,
                                const int*   __restrict__ shapes,
                                const int*   __restrict__ starts,
                                float* __restrict__ attn,
                                int Lq, int S) {
    const int wave = (int)((blockIdx.x * blockDim.x + threadIdx.x) >> 5);
    const int lane = threadIdx.x & 31;

    const int m  = wave & (N_HEADS - 1);
    const int nq = wave >> 3;            // n*Lq + q
    const int n  = nq / Lq;

    const size_t base16 = ((size_t)nq * N_HEADS + m) * (N_LEVELS * N_POINTS);
    const float* __restrict__ logit = awlog + base16;
    const float* __restrict__ offp  = off   + base16;

    // softmax over 16 logits (redundant per lane; cheap scalar work)
    float lg[N_LEVELS * N_POINTS];
    float mx = -3.4e38f;
#pragma unroll
    for (int i = 0; i < N_LEVELS * N_POINTS; ++i) {
        lg[i] = logit[i];
        mx = fmaxf(mx, lg[i]);
    }
    float sum = 0.0f;
#pragma unroll
    for (int i = 0; i < N_LEVELS * N_POINTS; ++i) {
        lg[i] = __expf(lg[i] - mx);
        sum += lg[i];
    }
    const float inv_sum = 1.0f / sum;

    const float* __restrict__ vbase = value + (size_t)n * S * D_MODEL + m * D_HEAD + lane;
    const float* __restrict__ rp    = refp  + (size_t)nq * N_LEVELS;

    float acc = 0.0f;
#pragma unroll
    for (int lp = 0; lp < N_LEVELS * N_POINTS; ++lp) {
        const int   l   = lp >> 2;
        const int   Ti  = shapes[l];
        const float T   = (float)Ti;
        const int   st  = starts[l];
        const float r   = rp[l];
        const float ov  = offp[lp];

        float x = r * T + ov - 0.5f;               // loc*T - 0.5
        x = fminf(fmaxf(x, 0.0f), T - 1.0f);
        const float x0 = floorf(x);
        const float w1 = x - x0;
        const int   i0 = (int)x0;
        const int   i1 = min(i0 + 1, Ti - 1);

        const float v0 = vbase[(size_t)(st + i0) * D_MODEL];
        const float v1 = vbase[(size_t)(st + i1) * D_MODEL];
        acc += (lg[lp] * inv_sum) * (v0 + w1 * (v1 - v0));
    }

    attn[(size_t)nq * D_MODEL + m * D_HEAD + lane] = acc;
}

// ---------------------------------------------------------------------------
extern "C" void kernel_launch(void* const* d_in, const int* in_sizes, int n_in,
                              void* d_out, int out_size, void* d_ws, size_t ws_size,
                              hipStream_t stream) {
    const float* query  = (const float*)d_in[0];
    const float* refp   = (const float*)d_in[1];
    const float* inpf   = (const float*)d_in[2];
    const int*   shapes = (const int*)d_in[3];
    const int*   starts = (const int*)d_in[4];
    const float* W_off  = (const float*)d_in[5];
    const float* b_off  = (const float*)d_in[6];
    const float* W_aw   = (const float*)d_in[7];
    const float* b_aw   = (const float*)d_in[8];
    const float* W_v    = (const float*)d_in[9];
    const float* b_v    = (const float*)d_in[10];
    const float* W_out  = (const float*)d_in[11];
    const float* b_out  = (const float*)d_in[12];
    float*       out    = (float*)d_out;

    const int Lq = in_sizes[0] / (NBATCH * D_MODEL);   // 4096
    const int S  = in_sizes[2] / (NBATCH * D_MODEL);   // 7680

    char* ws = (char*)d_ws;
    const size_t value_sz = (size_t)NBATCH * S  * D_MODEL * sizeof(float); // ~31.5 MB
    const size_t off_sz   = (size_t)NBATCH * Lq * HLP     * sizeof(float); // ~8.4 MB
    const size_t attn_sz  = (size_t)NBATCH * Lq * D_MODEL * sizeof(float); // ~16.8 MB
    const size_t w256_sz  = (size_t)D_MODEL * D_MODEL * sizeof(float);     // 256 KB
    const size_t w128_sz  = (size_t)D_MODEL * HLP     * sizeof(float);     // 128 KB

    char* p = ws;
    float* value = (float*)p;  p += value_sz;
    float* offb  = (float*)p;  p += off_sz;
    float* awb   = (float*)p;  p += off_sz;
    float* attn  = (float*)p;  p += attn_sz;
    float* WvT   = (float*)p;  p += w256_sz;
    float* WoutT = (float*)p;  p += w256_sz;
    float* WoffT = (float*)p;  p += w128_sz;
    float* WawT  = (float*)p;  p += w128_sz;

    // 0) one-time weight transposes (N x K layout for vectorized B loads)
    transpose_kn<D_MODEL, D_MODEL><<<(D_MODEL * D_MODEL) / 256, 256, 0, stream>>>(W_v,   WvT);
    transpose_kn<D_MODEL, D_MODEL><<<(D_MODEL * D_MODEL) / 256, 256, 0, stream>>>(W_out, WoutT);
    transpose_kn<D_MODEL, HLP    ><<<(D_MODEL * HLP    ) / 256, 256, 0, stream>>>(W_off, WoffT);
    transpose_kn<D_MODEL, HLP    ><<<(D_MODEL * HLP    ) / 256, 256, 0, stream>>>(W_aw,  WawT);

    // 1) value = input_flatten @ W_v + b_v      (30720 x 256, K=256)
    {
        dim3 grid((NBATCH * S) / 128, D_MODEL / 64);
        wmma_gemm_bias<D_MODEL, D_MODEL><<<grid, 128, 0, stream>>>(inpf, WvT, b_v, value);
    }
    // 2) off / aw logits = query @ {W_off, W_aw} + b    (16384 x 128, K=256)
    {
        dim3 grid((NBATCH * Lq) / 128, HLP / 64);
        wmma_gemm_bias<D_MODEL, HLP><<<grid, 128, 0, stream>>>(query, WoffT, b_off, offb);
        wmma_gemm_bias<D_MODEL, HLP><<<grid, 128, 0, stream>>>(query, WawT,  b_aw,  awb);
    }
    // 3) softmax + deformable linear sampling   (one wave per (n,q,m))
    {
        const int waves  = NBATCH * Lq * N_HEADS;      // 131072
        const int blocks = waves / 8;                  // 256 threads = 8 waves/block
        msdeform_sample<<<blocks, 256, 0, stream>>>(value, refp, offb, awb,
                                                    shapes, starts, attn, Lq, S);
    }
    // 4) out = attn @ W_out + b_out             (16384 x 256, K=256)
    {
        dim3 grid((NBATCH * Lq) / 128, D_MODEL / 64);
        wmma_gemm_bias<D_MODEL, D_MODEL><<<grid, 128, 0, stream>>>(attn, WoutT, b_out, out);
    }
}